// shooting_model_23433341567707
// MI455X (gfx1250) — compile-verified
//
#include <hip/hip_runtime.h>
#include <hip/hip_bf16.h>

// ---------------------------------------------------------------------------
// Shooting model for MI455X (gfx1250).
// B=16, H=W=512, L=10, fp32. Bandwidth-bound (~4.3 GB traffic total).
// Vertical Gaussian pass uses V_WMMA_F32_16X16X4_F32 as a banded matmul.
// Workspace requirement: 57 * 4194304 * 4 B  (~956 MB) from d_ws.
// ---------------------------------------------------------------------------

#define WW    512
#define HH    512
#define BB    16
#define NPIX  (HH * WW)          // 2^18
#define NTOT  (BB * NPIX)        // 4194304
#define LSTEP 10
#define MU2_OVER_L 0.00025f      // 0.05^2 / 10

typedef __attribute__((ext_vector_type(2))) float v2f;
typedef __attribute__((ext_vector_type(8))) float v8f;

// normalized 7-tap gaussian, sigma = 2
__device__ __constant__ float GW[7] = {
    0.07015933f, 0.13107488f, 0.19071282f, 0.21610594f,
    0.19071282f, 0.13107488f, 0.07015933f};

// zero-padded band table: GWPAD[i] = GW[i-16] for i in [16,22], else 0.
// Index d+16 with d = K - m in [-15, 23]  ->  always in [1, 39]: branchless.
__device__ __constant__ float GWPAD[40] = {
    0, 0, 0, 0, 0, 0, 0, 0, 0, 0, 0, 0, 0, 0, 0, 0,
    0.07015933f, 0.13107488f, 0.19071282f, 0.21610594f,
    0.19071282f, 0.13107488f, 0.07015933f,
    0, 0, 0, 0, 0, 0, 0, 0, 0, 0, 0, 0, 0, 0, 0, 0, 0};

// ---------------------------------------------------------------------------
// helpers: bilinear sample, zero padding, per-corner validity (matches ref)
// ---------------------------------------------------------------------------
__device__ __forceinline__ float g1(const float* __restrict__ fb, int xi, int yi) {
  return (xi >= 0 && xi < WW && yi >= 0 && yi < HH) ? fb[yi * WW + xi] : 0.0f;
}
__device__ __forceinline__ float bilin1(const float* __restrict__ fb, float sx, float sy) {
  float xf = floorf(sx), yf = floorf(sy);
  int   x0 = (int)xf,    y0 = (int)yf;
  float wx = sx - xf,    wy = sy - yf;
  float Ia = g1(fb, x0,     y0);
  float Ib = g1(fb, x0 + 1, y0);
  float Ic = g1(fb, x0,     y0 + 1);
  float Id = g1(fb, x0 + 1, y0 + 1);
  return Ia * (1.0f - wx) * (1.0f - wy) + Ib * wx * (1.0f - wy) +
         Ic * (1.0f - wx) * wy          + Id * wx * wy;
}
__device__ __forceinline__ float2 g2(const float2* __restrict__ fb, int xi, int yi) {
  if (xi >= 0 && xi < WW && yi >= 0 && yi < HH) return fb[yi * WW + xi];
  return make_float2(0.0f, 0.0f);
}
__device__ __forceinline__ float2 bilin2(const float2* __restrict__ fb, float sx, float sy) {
  float xf = floorf(sx), yf = floorf(sy);
  int   x0 = (int)xf,    y0 = (int)yf;
  float wx = sx - xf,    wy = sy - yf;
  float2 Ia = g2(fb, x0,     y0);
  float2 Ib = g2(fb, x0 + 1, y0);
  float2 Ic = g2(fb, x0,     y0 + 1);
  float2 Id = g2(fb, x0 + 1, y0 + 1);
  float wa = (1.0f - wx) * (1.0f - wy), wb = wx * (1.0f - wy);
  float wc = (1.0f - wx) * wy,          wd = wx * wy;
  return make_float2(Ia.x * wa + Ib.x * wb + Ic.x * wc + Id.x * wd,
                     Ia.y * wa + Ib.y * wb + Ic.y * wc + Id.y * wd);
}

// ---------------------------------------------------------------------------
// 1) Sobel spatial gradient (replicate pad) fused with p = -z * g
//    p layout: [B, 2, H, W] planar (dx plane, dy plane)
// ---------------------------------------------------------------------------
__global__ __launch_bounds__(256) void grad_premul(
    const float* __restrict__ img, const float* __restrict__ z,
    float* __restrict__ p) {
  int idx = blockIdx.x * blockDim.x + threadIdx.x;
  if (idx >= NTOT) return;
  int x = idx & (WW - 1);
  int y = (idx >> 9) & (HH - 1);
  int b = idx >> 18;
  const float* ib = img + (size_t)b * NPIX;
  int xm = max(x - 1, 0), xp = min(x + 1, WW - 1);
  int ym = max(y - 1, 0), yp = min(y + 1, HH - 1);
  float a00 = ib[ym * WW + xm], a01 = ib[ym * WW + x], a02 = ib[ym * WW + xp];
  float a10 = ib[y  * WW + xm],                        a12 = ib[y  * WW + xp];
  float a20 = ib[yp * WW + xm], a21 = ib[yp * WW + x], a22 = ib[yp * WW + xp];
  float gx = (a02 - a00 + 2.0f * (a12 - a10) + a22 - a20) * 0.125f;
  float gy = (a20 - a00 + 2.0f * (a21 - a01) + a22 - a02) * 0.125f;
  float zn = -z[idx];
  float* pb = p + (size_t)b * 2 * NPIX + y * WW + x;
  pb[0]    = zn * gx;
  pb[NPIX] = zn * gy;
}

// ---------------------------------------------------------------------------
// 2) horizontal 7-tap gaussian, zero pad.  planes = 2*B.
// ---------------------------------------------------------------------------
__global__ __launch_bounds__(256) void blur_h(
    const float* __restrict__ p, float* __restrict__ t) {
  int idx = blockIdx.x * blockDim.x + threadIdx.x;
  if (idx >= 2 * NTOT) return;
  int x = idx & (WW - 1);
  const float* row = p + (idx - x);
  float acc = 0.0f;
#pragma unroll
  for (int k = 0; k < 7; ++k) {
    int xx = x + k - 3;
    if (xx >= 0 && xx < WW) acc += GW[k] * row[xx];
  }
  t[idx] = acc;
}

// ---------------------------------------------------------------------------
// 3) vertical 7-tap gaussian via V_WMMA_F32_16X16X4_F32 banded matmul.
//    One wave per 16x16 output tile.
//    D[m,n] = sum_K A[m,K] * Bm[K,n],  A[m,K] = GW[K-m] for K-m in [0,6],
//    Bm[K,n] = t[y0-3+K, x0+n]  (zero outside, K padded 22->24).
//    A frag (16x4 f32): lane l: M=l%16, VGPR0=K(4c+2*(l/16)), VGPR1=K+1.
//    B frag (4x16 f32): lane l: N=l%16, VGPR0=row 4c+2*(l/16), VGPR1=row+1.
//    D (16x16 f32):     VGPR r: lanes0-15 M=r, lanes16-31 M=r+8  -> coalesced
//    row stores. Boundary check is wave-uniform (only ty==0 / ty==31 pad).
// ---------------------------------------------------------------------------
__global__ __launch_bounds__(256) void blur_v_wmma(
    const float* __restrict__ t, float* __restrict__ v) {
  int wave = (blockIdx.x * blockDim.x + threadIdx.x) >> 5;
  int lane = threadIdx.x & 31;
  int tx = wave & 31;           // 32 x-tiles
  int ty = (wave >> 5) & 31;    // 32 y-tiles
  int pl = wave >> 10;          // 32 planes (2*B)
  const float* tp = t + (size_t)pl * NPIX;
  float*       vp = v + (size_t)pl * NPIX;
  int x0 = tx << 4, y0 = ty << 4;
  int half = lane >> 4, ln = lane & 15;
  const float* col = tp + x0 + ln;

  // gather B-fragment rows (12 coalesced half-wave row-segment loads)
  float bx[6], by[6];
  if (y0 >= 3 && y0 + 20 < HH) {   // interior: wave-uniform fast path
#pragma unroll
    for (int c = 0; c < 6; ++c) {
      int yA = y0 - 3 + 4 * c + 2 * half;
      bx[c] = col[(size_t)yA * WW];
      by[c] = col[(size_t)(yA + 1) * WW];
    }
  } else {                         // top/bottom tiles: zero pad
#pragma unroll
    for (int c = 0; c < 6; ++c) {
      int yA = y0 - 3 + 4 * c + 2 * half;
      bx[c] = (yA     >= 0 && yA     < HH) ? col[(size_t)yA       * WW] : 0.0f;
      by[c] = (yA + 1 >= 0 && yA + 1 < HH) ? col[(size_t)(yA + 1) * WW] : 0.0f;
    }
  }

  v8f acc = {0.f, 0.f, 0.f, 0.f, 0.f, 0.f, 0.f, 0.f};
#pragma unroll
  for (int c = 0; c < 6; ++c) {
    int K0 = 4 * c + 2 * half;
    v2f a;                                    // branchless padded band table
    a.x = GWPAD[K0 - ln + 16];
    a.y = GWPAD[K0 - ln + 17];
    v2f bf;
    bf.x = bx[c];
    bf.y = by[c];
    acc = __builtin_amdgcn_wmma_f32_16x16x4_f32(
        /*neg_a=*/false, a, /*neg_b=*/false, bf,
        /*c_mod=*/(short)0, acc, /*reuse_a=*/false, /*reuse_b=*/false);
  }
  int yo = y0 + 8 * half;
#pragma unroll
  for (int r = 0; r < 8; ++r)
    vp[(size_t)(yo + r) * WW + x0 + ln] = acc[r];
}

// ---------------------------------------------------------------------------
// 4) fused:  f = div(z*v)/L (3x3, zero pad); z_next = z - f
//            phi_new = id_grid - v/L  ([B,H,W,2] interleaved)
// ---------------------------------------------------------------------------
__global__ __launch_bounds__(256) void div_res_deform(
    const float* __restrict__ zi, const float* __restrict__ v,
    float* __restrict__ znext, float* __restrict__ phi) {
  int idx = blockIdx.x * blockDim.x + threadIdx.x;
  if (idx >= NTOT) return;
  int x = idx & (WW - 1);
  int y = (idx >> 9) & (HH - 1);
  int b = idx >> 18;
  const float* vx = v + (size_t)b * 2 * NPIX;
  const float* vy = vx + NPIX;
  const float* zb = zi + (size_t)b * NPIX;
  // sobel_x (ch0) and sobel_x^T (ch1), un-normalized; 1/8 folded below
  const float k0[3][3] = {{-1.f, 0.f, 1.f}, {-2.f, 0.f, 2.f}, {-1.f, 0.f, 1.f}};
  const float k1[3][3] = {{-1.f, -2.f, -1.f}, {0.f, 0.f, 0.f}, {1.f, 2.f, 1.f}};
  float acc = 0.0f;
#pragma unroll
  for (int dy = 0; dy < 3; ++dy) {
#pragma unroll
    for (int dx = 0; dx < 3; ++dx) {
      int yy = y + dy - 1, xx = x + dx - 1;
      if (yy >= 0 && yy < HH && xx >= 0 && xx < WW) {
        int o = yy * WW + xx;
        float zz = zb[o];
        acc += k0[dy][dx] * (zz * vx[o]) + k1[dy][dx] * (zz * vy[o]);
      }
    }
  }
  znext[idx] = zi[idx] - acc * (0.125f / (float)LSTEP);
  int oc = y * WW + x;
  ((float2*)phi)[idx] = make_float2((float)x - vx[oc] * (1.0f / LSTEP),
                                    (float)y - vy[oc] * (1.0f / LSTEP));
}

// ---------------------------------------------------------------------------
// 5) warp all previous phi fields through new deformation (blockIdx.y = j)
// ---------------------------------------------------------------------------
__global__ __launch_bounds__(256) void warp_phi(
    const float* __restrict__ oldset, const float* __restrict__ defo,
    float* __restrict__ newset) {
  int j   = blockIdx.y;
  int idx = blockIdx.x * blockDim.x + threadIdx.x;
  if (idx >= NTOT) return;
  int b = idx >> 18;
  float2 d = ((const float2*)defo)[idx];
  const float2* of = (const float2*)oldset + (size_t)j * NTOT + (size_t)b * NPIX;
  float2*       nf = (float2*)newset       + (size_t)j * NTOT;
  nf[idx] = bilin2(of, d.x, d.y);
}

// ---------------------------------------------------------------------------
// 6) fused residual sum + image update:
//    rsum  = z_{i+1} + sum_{j=1..i} grid_sample(z_j, phi_j)
//    mask  = sample(seg, phi_0)
//    image = sample(src, phi_0) + rsum * (mu^2/L) * mask
//    zbuf[k] holds z_{k+1}
// ---------------------------------------------------------------------------
__global__ __launch_bounds__(256) void update_image(
    const float* __restrict__ src, const float* __restrict__ seg,
    const float* __restrict__ phiset, const float* __restrict__ zbuf,
    const float* __restrict__ znext, float* __restrict__ out, int i) {
  int idx = blockIdx.x * blockDim.x + threadIdx.x;
  if (idx >= NTOT) return;
  int b = idx >> 18;
  float acc = znext[idx];
  for (int j = 1; j <= i; ++j) {
    float2 d = ((const float2*)phiset + (size_t)j * NTOT)[idx];
    const float* zj = zbuf + (size_t)(j - 1) * NTOT + (size_t)b * NPIX;
    acc += bilin1(zj, d.x, d.y);
  }
  float2 d0 = ((const float2*)phiset)[idx];
  float m = bilin1(seg + (size_t)b * NPIX, d0.x, d0.y);
  float s = bilin1(src + (size_t)b * NPIX, d0.x, d0.y);
  out[idx] = s + acc * MU2_OVER_L * m;
}

// ---------------------------------------------------------------------------
// host side
// ---------------------------------------------------------------------------
extern "C" void kernel_launch(void* const* d_in, const int* in_sizes, int n_in,
                              void* d_out, int out_size, void* d_ws,
                              size_t ws_size, hipStream_t stream) {
  const float* source = (const float*)d_in[0];  // [B,1,H,W]
  const float* z0in   = (const float*)d_in[1];  // [B,1,H,W]
  const float* seg    = (const float*)d_in[2];  // [B,1,H,W]
  float* out = (float*)d_out;

  const size_t N = (size_t)NTOT;
  float* ws   = (float*)d_ws;
  float* img  = ws;  ws += N;            // current image
  float* p    = ws;  ws += 2 * N;        // -z * gradient      [B,2,H,W]
  float* tbuf = ws;  ws += 2 * N;        // h-blur intermediate
  float* vbuf = ws;  ws += 2 * N;        // velocity field     [B,2,H,W]
  float* zbuf = ws;  ws += (size_t)LSTEP * N;       // z_1 .. z_L
  float* phiA = ws;  ws += (size_t)LSTEP * 2 * N;   // phi set (old)
  float* phiB = ws;  /* += LSTEP*2*N */              // phi set (new)

  const int TPB = 256;
  const int gN = (NTOT + TPB - 1) / TPB;  // 16384 blocks

  const float* cur_img = source;
  const float* cur_z   = z0in;

  for (int i = 0; i < LSTEP; ++i) {
    float* znext = zbuf + (size_t)i * N;           // z_{i+1}
    float* newD  = phiB + (size_t)i * 2 * N;       // new deformation slot

    grad_premul<<<gN, TPB, 0, stream>>>(cur_img, cur_z, p);
    blur_h<<<2 * gN, TPB, 0, stream>>>(p, tbuf);
    blur_v_wmma<<<4096, TPB, 0, stream>>>(tbuf, vbuf);   // 32768 waves, 1 tile each
    div_res_deform<<<gN, TPB, 0, stream>>>(cur_z, vbuf, znext, newD);
    if (i > 0) {
      dim3 grid(gN, i);
      warp_phi<<<grid, TPB, 0, stream>>>(phiA, newD, phiB);
    }
    float* img_out = (i == LSTEP - 1) ? out : img;
    update_image<<<gN, TPB, 0, stream>>>(source, seg, phiB, zbuf, znext,
                                         img_out, i);

    cur_img = img_out;
    cur_z   = znext;
    float* sw = phiA; phiA = phiB; phiB = sw;  // ping-pong phi sets
  }
}